// OnnxScatterND_13950053778326
// MI455X (gfx1250) — compile-verified
//
#include <hip/hip_runtime.h>
#include <cstdint>
#include <cstddef>

// 128-bit vector type for global_load_b128 / global_store_b128 with
// non-temporal cache policy (streaming, single-touch > L2 working set).
typedef __attribute__((ext_vector_type(4))) float v4f;

// ---------------------------------------------------------------------------
// Kernel 1: zero the per-row "winner" table (d_ws is poisoned by the harness,
// so this must run every call; it is tiny: n_rows * 4 bytes = 2 MB).
// ---------------------------------------------------------------------------
__global__ __launch_bounds__(256) void init_winner(unsigned* __restrict__ w,
                                                   int n_rows) {
    int i = blockIdx.x * blockDim.x + threadIdx.x;
    if (i < n_rows) w[i] = 0u;
}

// ---------------------------------------------------------------------------
// Kernel 2: deterministic last-write-wins via atomicMax over update index.
// winner[row] = 1 + (largest update index targeting row), or 0 if untouched.
// 200 000 atomics on a 2 MB table -> negligible vs. the 1 GB stream.
// ---------------------------------------------------------------------------
__global__ __launch_bounds__(256) void winner_pass(const int64_t* __restrict__ idx,
                                                   unsigned* __restrict__ w,
                                                   int n_upd, int n_rows) {
    int i = blockIdx.x * blockDim.x + threadIdx.x;
    if (i >= n_upd) return;
    int64_t r = idx[i];
    if (r >= 0 && r < (int64_t)n_rows) {
        atomicMax(&w[(int)r], (unsigned)(i + 1));
    }
}

// ---------------------------------------------------------------------------
// Kernel 3: gather-formulated scatter. One thread per 16-byte chunk of the
// output. A row (256 floats) = 64 chunks, so 64 consecutive threads (two
// full wave32s) own one row -> the winner test is wave-uniform (no lane
// divergence) and both load & store streams are perfectly coalesced B128.
// Non-temporal hints keep the 512 MB write + ~512 MB reads from evicting L2.
// ---------------------------------------------------------------------------
__global__ __launch_bounds__(256) void gather_rows(const float* __restrict__ data,
                                                   const float* __restrict__ upd,
                                                   const unsigned* __restrict__ w,
                                                   float* __restrict__ out,
                                                   unsigned vec_per_row,  // cols/4
                                                   unsigned total_vec) {  // n_rows*vec_per_row
    unsigned gid = blockIdx.x * blockDim.x + threadIdx.x;
    if (gid >= total_vec) return;
    unsigned row = gid / vec_per_row;
    unsigned c   = gid % vec_per_row;

    unsigned win = w[row];  // cached; 64 threads share one entry

    size_t row_elems = (size_t)vec_per_row * 4u;
    const float* src_row = win ? (upd  + (size_t)(win - 1u) * row_elems)
                               : (data + (size_t)row        * row_elems);

    v4f v = __builtin_nontemporal_load((const v4f*)src_row + c);
    __builtin_nontemporal_store(v, (v4f*)(out + (size_t)row * row_elems) + c);
}

// ---------------------------------------------------------------------------
// Launch: inputs are (data f32[500000,256], indices i64[200000,1],
// updates f32[200000,256]); output f32[500000,256]. All sizes derived from
// in_sizes so the code stays shape-generic.
// ---------------------------------------------------------------------------
extern "C" void kernel_launch(void* const* d_in, const int* in_sizes, int n_in,
                              void* d_out, int out_size, void* d_ws, size_t ws_size,
                              hipStream_t stream) {
    const float*   data    = (const float*)d_in[0];
    const int64_t* indices = (const int64_t*)d_in[1];
    const float*   updates = (const float*)d_in[2];
    float*         out     = (float*)d_out;

    const int n_upd  = in_sizes[1];              // 200000 (k = 1)
    const int cols   = in_sizes[2] / n_upd;      // 256
    const int n_rows = in_sizes[0] / cols;       // 500000

    unsigned* winner = (unsigned*)d_ws;          // n_rows * 4 bytes of scratch

    {
        int threads = 256;
        int blocks  = (n_rows + threads - 1) / threads;
        init_winner<<<blocks, threads, 0, stream>>>(winner, n_rows);
    }
    {
        int threads = 256;
        int blocks  = (n_upd + threads - 1) / threads;
        winner_pass<<<blocks, threads, 0, stream>>>(indices, winner, n_upd, n_rows);
    }
    {
        const unsigned vec_per_row = (unsigned)(cols / 4);            // 64
        const unsigned total_vec   = (unsigned)n_rows * vec_per_row;  // 32M
        int threads = 256;
        unsigned blocks = (total_vec + (unsigned)threads - 1) / (unsigned)threads;
        gather_rows<<<blocks, threads, 0, stream>>>(data, updates, winner, out,
                                                    vec_per_row, total_vec);
    }
}